// pyramid_step_model_1468878815900
// MI455X (gfx1250) — compile-verified
//
#include <hip/hip_runtime.h>
#include <math.h>

#define Bv    4
#define Tv    4096
#define Sv    4096
#define Ev    16
#define NHv   16
#define Hv    4
#define HDv   4
#define MDv   256
#define FFv   1024
#define FFNHv 64
#define OUTv  16

typedef __attribute__((ext_vector_type(2))) float v2f;
typedef __attribute__((ext_vector_type(8))) float v8f;

__device__ __forceinline__ float gelu_exact(float x) {
    // torch nn.GELU default: 0.5*x*(1+erf(x/sqrt(2)))
    return 0.5f * x * (1.0f + erff(x * 0.70710678118654752440f));
}

// ---------------------------------------------------------------------------
// Kernel 1: exact top-16 nearest source neighbors per target.
// One thread per target; coords_source streamed through LDS in 256-pt tiles.
// Register-resident sorted insertion chain; strict '<' matches lax.top_k tie
// order (earlier index wins on equal distance).
// ---------------------------------------------------------------------------
__global__ void __launch_bounds__(256) knn_kernel(const float* __restrict__ ct,
                                                  const float* __restrict__ cs,
                                                  int* __restrict__ idx_out) {
    __shared__ float2 tile[256];
    int gid = blockIdx.x * 256 + threadIdx.x;   // b*T + t, block never straddles b
    int b = gid >> 12;                          // T = 4096
    float tx = ct[(size_t)gid * 2 + 0];
    float ty = ct[(size_t)gid * 2 + 1];

    float bestd[16];
    int   besti[16];
#pragma unroll
    for (int q = 0; q < 16; ++q) { bestd[q] = 3.0e38f; besti[q] = 0; }

    const float2* cs2 = (const float2*)cs + (size_t)b * Sv;
    for (int s0 = 0; s0 < Sv; s0 += 256) {
        __syncthreads();
        tile[threadIdx.x] = cs2[s0 + threadIdx.x];
        __syncthreads();
        for (int s = 0; s < 256; ++s) {
            float2 c = tile[s];
            float dx = c.x - tx, dy = c.y - ty;
            float d = dx * dx + dy * dy;
            if (d < bestd[15]) {
                float dd = d;
                int   id = s0 + s;
#pragma unroll
                for (int q = 0; q < 16; ++q) {
                    bool  sw = dd < bestd[q];
                    float td = sw ? bestd[q] : dd;
                    int   ti = sw ? besti[q] : id;
                    bestd[q] = sw ? dd : bestd[q];
                    besti[q] = sw ? id : besti[q];
                    dd = td; id = ti;
                }
            }
        }
    }
#pragma unroll
    for (int q = 0; q < 16; ++q) idx_out[(size_t)gid * 16 + q] = besti[q];
}

// ---------------------------------------------------------------------------
// Kernel 2: gather + PE-MLP + LayerNorm + q/k/v proj + cosine local attention
// + out-proj + per-neighbor MLP. 16 threads per target (one per neighbor);
// 16 targets per 256-thread block. Small weights staged in LDS once.
// Writes the reassembled [B*T, 256] activation row-major to workspace.
// ---------------------------------------------------------------------------
__global__ void __launch_bounds__(256) attn_kernel(
    const float* __restrict__ x, const float* __restrict__ ct, const float* __restrict__ cs,
    const int* __restrict__ idx,
    const float* __restrict__ pe_w1, const float* __restrict__ pe_b1,
    const float* __restrict__ pe_w2, const float* __restrict__ pe_b2,
    const float* __restrict__ ln_g, const float* __restrict__ ln_b,
    const float* __restrict__ q_w, const float* __restrict__ k_w, const float* __restrict__ v_w,
    const float* __restrict__ logit_scale, const float* __restrict__ out_w,
    const float* __restrict__ mnh_w1, const float* __restrict__ mnh_b1,
    const float* __restrict__ mnh_w2, const float* __restrict__ mnh_b2,
    float* __restrict__ o_md) {

    __shared__ float w_pe1[128], w_pe1b[64], w_pe2[1024], w_pe2b[16];
    __shared__ float w_lng[16], w_lnb[16];
    __shared__ float w_q[256], w_k[256], w_v[256], w_o[256];
    __shared__ float w_ls[4];
    __shared__ float w_m1[1024], w_m1b[64], w_m2[64], w_m2b[1];
    __shared__ float sh_kn[16][16][16];   // [target][neighbor][dim] normalized k
    __shared__ float sh_v [16][16][16];   // [target][neighbor][dim] v

    int tid = threadIdx.x;
    for (int i = tid; i < 128;  i += 256) w_pe1[i] = pe_w1[i];
    for (int i = tid; i < 64;   i += 256) w_pe1b[i] = pe_b1[i];
    for (int i = tid; i < 1024; i += 256) w_pe2[i] = pe_w2[i];
    for (int i = tid; i < 1024; i += 256) w_m1[i]  = mnh_w1[i];
    for (int i = tid; i < 64;   i += 256) { w_m1b[i] = mnh_b1[i]; w_m2[i] = mnh_w2[i]; }
    if (tid < 16) { w_pe2b[tid] = pe_b2[tid]; w_lng[tid] = ln_g[tid]; w_lnb[tid] = ln_b[tid]; }
    if (tid < 256) { w_q[tid] = q_w[tid]; w_k[tid] = k_w[tid]; w_v[tid] = v_w[tid]; w_o[tid] = out_w[tid]; }
    if (tid < 4)  w_ls[tid]  = logit_scale[tid];
    if (tid == 0) w_m2b[0]   = mnh_b2[0];
    __syncthreads();

    int tl = tid >> 4, nb = tid & 15;
    int g = blockIdx.x * 16 + tl;       // b*T + t
    int b = g >> 12;

    int j = idx[(size_t)g * 16 + nb];
    float relx = cs[((size_t)b * Sv + j) * 2 + 0] - ct[(size_t)g * 2 + 0];
    float rely = cs[((size_t)b * Sv + j) * 2 + 1] - ct[(size_t)g * 2 + 1];
    float xrow[16];
#pragma unroll
    for (int e = 0; e < 16; ++e) xrow[e] = x[((size_t)b * Sv + j) * 16 + e];

    // relative position embedder: gelu(rel @ pe_w1 + b1) @ pe_w2 + b2
    float h[16];
#pragma unroll
    for (int d = 0; d < 16; ++d) h[d] = w_pe2b[d];
    for (int f = 0; f < 64; ++f) {
        float a = gelu_exact(relx * w_pe1[f] + rely * w_pe1[64 + f] + w_pe1b[f]);
#pragma unroll
        for (int d = 0; d < 16; ++d) h[d] += a * w_pe2[f * 16 + d];
    }
#pragma unroll
    for (int d = 0; d < 16; ++d) h[d] += xrow[d];

    // LayerNorm over last dim 16 (population variance, eps 1e-5)
    float mu = 0.f;
#pragma unroll
    for (int d = 0; d < 16; ++d) mu += h[d];
    mu *= (1.f / 16.f);
    float var = 0.f;
#pragma unroll
    for (int d = 0; d < 16; ++d) { float c = h[d] - mu; var += c * c; }
    var *= (1.f / 16.f);
    float inv = 1.0f / sqrtf(var + 1e-5f);
#pragma unroll
    for (int d = 0; d < 16; ++d) h[d] = (h[d] - mu) * inv * w_lng[d] + w_lnb[d];

    // q/k/v projections (q,k from normed h; v from x_nh)
    float q[16], kreg[16];
#pragma unroll
    for (int d = 0; d < 16; ++d) {
        float aq = 0.f, ak = 0.f, av = 0.f;
#pragma unroll
        for (int e = 0; e < 16; ++e) {
            aq += h[e]    * w_q[e * 16 + d];
            ak += h[e]    * w_k[e * 16 + d];
            av += xrow[e] * w_v[e * 16 + d];
        }
        q[d] = aq; kreg[d] = ak;
        sh_v[tl][nb][d] = av;
    }
    // per-head L2 normalization (cosine attention), eps 1e-6
#pragma unroll
    for (int hh = 0; hh < 4; ++hh) {
        float nq = 0.f, nk = 0.f;
#pragma unroll
        for (int d = 0; d < 4; ++d) {
            nq += q[hh * 4 + d] * q[hh * 4 + d];
            nk += kreg[hh * 4 + d] * kreg[hh * 4 + d];
        }
        float iq = 1.f / (sqrtf(nq) + 1e-6f);
        float ik = 1.f / (sqrtf(nk) + 1e-6f);
#pragma unroll
        for (int d = 0; d < 4; ++d) {
            q[hh * 4 + d] *= iq;
            sh_kn[tl][nb][hh * 4 + d] = kreg[hh * 4 + d] * ik;
        }
    }
    __syncthreads();

    // local attention: this thread = query neighbor i = nb
    float ofin[16];
#pragma unroll
    for (int d = 0; d < 16; ++d) ofin[d] = 0.f;
#pragma unroll
    for (int hh = 0; hh < 4; ++hh) {
        float lsc = expf(fminf(w_ls[hh], 4.60517018598809136804f));  // min(ls, log 100)
        float lg[16];
        float mx = -3.0e38f;
#pragma unroll
        for (int jj = 0; jj < 16; ++jj) {
            float dot = 0.f;
#pragma unroll
            for (int d = 0; d < 4; ++d) dot += q[hh * 4 + d] * sh_kn[tl][jj][hh * 4 + d];
            lg[jj] = dot * lsc;
            mx = fmaxf(mx, lg[jj]);
        }
        float se = 0.f;
#pragma unroll
        for (int jj = 0; jj < 16; ++jj) { lg[jj] = expf(lg[jj] - mx); se += lg[jj]; }
        float isum = 1.f / se;
#pragma unroll
        for (int jj = 0; jj < 16; ++jj) {
            float a = lg[jj] * isum;
#pragma unroll
            for (int d = 0; d < 4; ++d) ofin[hh * 4 + d] += a * sh_v[tl][jj][hh * 4 + d];
        }
    }

    // attention out-proj
    float oo[16];
#pragma unroll
    for (int d = 0; d < 16; ++d) {
        float a = 0.f;
#pragma unroll
        for (int e = 0; e < 16; ++e) a += ofin[e] * w_o[e * 16 + d];
        oo[d] = a;
    }
    // per-neighbor MLP (16 -> 64 -> 1), broadcast residual add
    float s1 = 0.f;
    for (int f = 0; f < 64; ++f) {
        float a = w_m1b[f];
#pragma unroll
        for (int e = 0; e < 16; ++e) a += oo[e] * w_m1[e * 64 + f];
        s1 += gelu_exact(a) * w_m2[f];
    }
    float m = gelu_exact(s1 + w_m2b[0]);
#pragma unroll
    for (int d = 0; d < 16; ++d) o_md[(size_t)g * 256 + nb * 16 + d] = oo[d] + m;
}

// ---------------------------------------------------------------------------
// Kernel 3: fused output MLP with fp32 WMMA (V_WMMA_F32_16X16X4_F32).
//   out = gelu( gelu( A[16384,256] @ W1[256,1024] + b1 ) @ W2[1024,16] + b2 )
// One block per 16-row tile. 8 waves: each computes 8 N-tiles of GEMM1
// (64 wmma k-steps each, C preloaded with b1), GELUs in-register, bounces the
// 16x16 Z tile through LDS to re-lay out D->A, then accumulates its K-slice of
// GEMM2 with 4 more wmma steps per tile. Cross-wave reduce + bias + GELU.
// ---------------------------------------------------------------------------
__global__ void __launch_bounds__(256) mlp_kernel(
    const float* __restrict__ A, const float* __restrict__ W1, const float* __restrict__ B1,
    const float* __restrict__ W2, const float* __restrict__ B2, float* __restrict__ out) {

    __shared__ float sA[16][257];       // padded: stride 257 -> conflict-free column reads
    __shared__ float sZ[8][16][17];     // per-wave D->A relayout buffer
    __shared__ float sRed[8][16][16];   // cross-wave partial tiles

    int tid   = threadIdx.x;
    int wave  = tid >> 5;
    int lane  = tid & 31;
    int lhalf = lane & 15;
    bool hiu  = lane >= 16;
    int m0    = blockIdx.x * 16;

    for (int i = tid; i < 16 * 256; i += 256)
        sA[i >> 8][i & 255] = A[(size_t)(m0 + (i >> 8)) * 256 + (i & 255)];
    __syncthreads();

    v8f acc2;
#pragma unroll
    for (int r = 0; r < 8; ++r) acc2[r] = 0.f;

    int koff  = hiu ? 2 : 0;
    int rbase = hiu ? 8 : 0;

    for (int nt = 0; nt < 8; ++nt) {
        int n0 = (wave * 8 + nt) * 16;

        // C preloaded with bias b1 (C layout: lane -> column N, VGPR r -> row)
        float bb = B1[n0 + lhalf];
        v8f acc;
#pragma unroll
        for (int r = 0; r < 8; ++r) acc[r] = bb;

        // GEMM1: K = 256 in steps of 4
        for (int kk = 0; kk < 256; kk += 4) {
            int ka = kk + koff;
            v2f a;  a.x = sA[lhalf][ka];                       // A[m=lhalf][ka]
                    a.y = sA[lhalf][ka + 1];
            v2f bv; bv.x = W1[(size_t)ka * 1024 + n0 + lhalf]; // B[k=ka][n=n0+lhalf]
                    bv.y = W1[(size_t)(ka + 1) * 1024 + n0 + lhalf];
            acc = __builtin_amdgcn_wmma_f32_16x16x4_f32(false, a, false, bv,
                                                        (short)0, acc, false, false);
        }
        // elementwise GELU in C layout
#pragma unroll
        for (int r = 0; r < 8; ++r) acc[r] = gelu_exact(acc[r]);

        // bounce D layout (lane=N, vgpr=M) -> A layout (lane=M, vgpr=K) via LDS
#pragma unroll
        for (int r = 0; r < 8; ++r) sZ[wave][rbase + r][lhalf] = acc[r];
        // per-wave LDS RAW; hardware keeps DS ops in order, compiler waits DScnt

        // GEMM2 partial: this Z tile covers K-range [n0, n0+16) of the 1024-K GEMM
#pragma unroll
        for (int kk2 = 0; kk2 < 16; kk2 += 4) {
            int ka = kk2 + koff;
            v2f a2;  a2.x = sZ[wave][lhalf][ka];
                     a2.y = sZ[wave][lhalf][ka + 1];
            v2f b2v; b2v.x = W2[(size_t)(n0 + ka) * 16 + lhalf];
                     b2v.y = W2[(size_t)(n0 + ka + 1) * 16 + lhalf];
            acc2 = __builtin_amdgcn_wmma_f32_16x16x4_f32(false, a2, false, b2v,
                                                         (short)0, acc2, false, false);
        }
    }

    // cross-wave reduction of the 8 K-slice partials
#pragma unroll
    for (int r = 0; r < 8; ++r) sRed[wave][rbase + r][lhalf] = acc2[r];
    __syncthreads();

    int m = tid >> 4, n = tid & 15;
    float s = B2[n];
#pragma unroll
    for (int w = 0; w < 8; ++w) s += sRed[w][m][n];
    out[(size_t)(m0 + m) * 16 + n] = gelu_exact(s);
}

// ---------------------------------------------------------------------------
extern "C" void kernel_launch(void* const* d_in, const int* in_sizes, int n_in,
                              void* d_out, int out_size, void* d_ws, size_t ws_size,
                              hipStream_t stream) {
    const float* x        = (const float*)d_in[0];
    const float* ct       = (const float*)d_in[1];
    const float* cs       = (const float*)d_in[2];
    const float* pe_w1    = (const float*)d_in[3];
    const float* pe_b1    = (const float*)d_in[4];
    const float* pe_w2    = (const float*)d_in[5];
    const float* pe_b2    = (const float*)d_in[6];
    const float* ln_g     = (const float*)d_in[7];
    const float* ln_b     = (const float*)d_in[8];
    const float* q_w      = (const float*)d_in[9];
    const float* k_w      = (const float*)d_in[10];
    const float* v_w      = (const float*)d_in[11];
    const float* lsc      = (const float*)d_in[12];
    const float* out_w    = (const float*)d_in[13];
    const float* mnh_w1   = (const float*)d_in[14];
    const float* mnh_b1   = (const float*)d_in[15];
    const float* mnh_w2   = (const float*)d_in[16];
    const float* mnh_b2   = (const float*)d_in[17];
    const float* mo_w1    = (const float*)d_in[18];
    const float* mo_b1    = (const float*)d_in[19];
    const float* mo_w2    = (const float*)d_in[20];
    const float* mo_b2    = (const float*)d_in[21];

    // workspace layout: [0, 1MB) int idx[B*T*16]; [1MB, 17MB) float o_md[B*T*256]
    int*   idx  = (int*)d_ws;
    float* o_md = (float*)((char*)d_ws + (1 << 20));

    const int BT = Bv * Tv;                       // 16384
    knn_kernel <<<BT / 256, 256, 0, stream>>>(ct, cs, idx);
    attn_kernel<<<BT / 16,  256, 0, stream>>>(x, ct, cs, idx,
                                              pe_w1, pe_b1, pe_w2, pe_b2,
                                              ln_g, ln_b, q_w, k_w, v_w,
                                              lsc, out_w,
                                              mnh_w1, mnh_b1, mnh_w2, mnh_b2,
                                              o_md);
    mlp_kernel <<<BT / 16,  256, 0, stream>>>(o_md, mo_w1, mo_b1, mo_w2, mo_b2,
                                              (float*)d_out);
}